// Encoder_31430570672733
// MI455X (gfx1250) — compile-verified
//
#include <hip/hip_runtime.h>

#define C_COMB 2048   // 2^(INPUT_DIM+1) table entries

#if __has_builtin(__builtin_amdgcn_global_load_async_to_lds_b32)
#define HAS_ASYNC_LDS 1
#else
#define HAS_ASYNC_LDS 0
#endif

typedef __attribute__((address_space(1))) int* gptr_i32;
typedef __attribute__((address_space(3))) int* lptr_i32;

__global__ __launch_bounds__(256) void encoder_gather_kernel(
    const float* __restrict__ x, const float* __restrict__ t,
    const float* __restrict__ y, const float* __restrict__ w,
    const float* __restrict__ b, float* __restrict__ out, int n)
{
    // Interleaved {w[i], b[i]} pairs: one ds_load_b64 per row gather. 16 KB.
    __shared__ float2 wb[C_COMB];

    // ---- Stage the table into LDS (CDNA5 async global->LDS path) ----
    for (int i = threadIdx.x; i < C_COMB; i += 256) {
#if HAS_ASYNC_LDS
        __builtin_amdgcn_global_load_async_to_lds_b32(
            (gptr_i32)(w + i), (lptr_i32)(&wb[i].x), 0, 0);
        __builtin_amdgcn_global_load_async_to_lds_b32(
            (gptr_i32)(b + i), (lptr_i32)(&wb[i].y), 0, 0);
#else
        wb[i].x = w[i];
        wb[i].y = b[i];
#endif
    }
#if HAS_ASYNC_LDS
    asm volatile("s_wait_asynccnt 0" ::: "memory");
#endif
    __syncthreads();

    const int gtid   = (int)(blockIdx.x * blockDim.x + threadIdx.x);
    const int stride = (int)(gridDim.x * blockDim.x);

    // ---- Two rows per thread: 80B x-window is 16B-aligned -> 5x b128 ----
    const int npair = n >> 1;
    for (int p = gtid; p < npair; p += stride) {
        const float4* xr = (const float4*)(x + (size_t)p * 20u);

        // Prefetch next grid-stride x window (speculative, OOB-safe).
        __builtin_prefetch(x + (size_t)(p + stride) * 20u, 0, 3);

        float4 a = xr[0];                        // r0: x0..x3
        float4 c = xr[1];                        // r0: x4..x7
        float4 e = xr[2];                        // r0: x8,x9 | r1: x0,x1
        float4 g = xr[3];                        // r1: x2..x5
        float4 h = xr[4];                        // r1: x6..x9
        float2 tv = *(const float2*)(t + 2 * (size_t)p);
        float2 yv = *(const float2*)(y + 2 * (size_t)p);

        // idx = t*2^10 + x0*2^9 + ... + x9 ; exact in f32 (max 2047)
        float f0 = tv.x;
        f0 = fmaf(f0, 2.0f, a.x); f0 = fmaf(f0, 2.0f, a.y);
        f0 = fmaf(f0, 2.0f, a.z); f0 = fmaf(f0, 2.0f, a.w);
        f0 = fmaf(f0, 2.0f, c.x); f0 = fmaf(f0, 2.0f, c.y);
        f0 = fmaf(f0, 2.0f, c.z); f0 = fmaf(f0, 2.0f, c.w);
        f0 = fmaf(f0, 2.0f, e.x); f0 = fmaf(f0, 2.0f, e.y);

        float f1 = tv.y;
        f1 = fmaf(f1, 2.0f, e.z); f1 = fmaf(f1, 2.0f, e.w);
        f1 = fmaf(f1, 2.0f, g.x); f1 = fmaf(f1, 2.0f, g.y);
        f1 = fmaf(f1, 2.0f, g.z); f1 = fmaf(f1, 2.0f, g.w);
        f1 = fmaf(f1, 2.0f, h.x); f1 = fmaf(f1, 2.0f, h.y);
        f1 = fmaf(f1, 2.0f, h.z); f1 = fmaf(f1, 2.0f, h.w);

        float2 v0 = wb[(int)f0];                 // ds_load_b64 gathers
        float2 v1 = wb[(int)f1];

        float2 o;
        o.x = fmaf(yv.x, v0.x, v0.y);
        o.y = fmaf(yv.y, v1.x, v1.y);
        *(float2*)(out + 2 * (size_t)p) = o;     // coalesced b64 store
    }

    // ---- Odd-n tail: one thread handles the last row ----
    if ((n & 1) && gtid == 0) {
        const int r = n - 1;
        const float* xr = x + (size_t)r * 10u;
        float f = t[r];
        #pragma unroll
        for (int j = 0; j < 10; ++j) f = fmaf(f, 2.0f, xr[j]);
        float2 v = wb[(int)f];
        out[r] = fmaf(y[r], v.x, v.y);
    }
}

extern "C" void kernel_launch(void* const* d_in, const int* in_sizes, int n_in,
                              void* d_out, int out_size, void* d_ws, size_t ws_size,
                              hipStream_t stream) {
    const float* x = (const float*)d_in[0];   // [N,10]
    const float* t = (const float*)d_in[1];   // [N,1]
    const float* y = (const float*)d_in[2];   // [N,1]
    const float* w = (const float*)d_in[3];   // [2048]
    const float* b = (const float*)d_in[4];   // [2048]
    float* out = (float*)d_out;               // [N,1]

    const int n = in_sizes[1];                // N rows (t has N elements)
    const int block = 256;                    // 8 wave32 waves
    int blocks = ((n >> 1) + block - 1) / block;
    if (blocks > 2048) blocks = 2048;         // persistent grid: cap table preloads
    if (blocks < 1) blocks = 1;

    encoder_gather_kernel<<<blocks, block, 0, stream>>>(x, t, y, w, b, out, n);
}